// StateSpaceDiscreteRK4_69552700391538
// MI455X (gfx1250) — compile-verified
//
#include <hip/hip_runtime.h>
#include <hip/hip_bf16.h>

// StateSpaceDiscreteRK4 for MI455X (gfx1250, wave32).
//
// Shapes: x(1024,4096) state(1024,4096,16) A/B/C(4096,16) D/dt_proj(4096)
// Out: y(1024,4096) ++ state passthrough copy (1024,4096,16), all fp32.
//
// Memory-bound streaming kernel (~545 MB => ~23us floor at 23.3 TB/s).
// WMMA f32 16x16x4 used (correctly) for the n=16 contraction: fold C
// elementwise, then row-sum via D = A x Ones (layout-invariant B).

typedef float v2f __attribute__((ext_vector_type(2)));
typedef float v4f __attribute__((ext_vector_type(4)));
typedef float v8f __attribute__((ext_vector_type(8)));

#define SSM_BATCH   1024
#define SSM_DINNER  4096
#define SSM_DSTATE  16
#define SSM_DTILE   16      // d-channels per block (= WMMA M rows)
#define SSM_BWAVES  8       // batches per block (one wave per batch)
#define SSM_LSTRIDE 17      // LDS padding stride (bank-conflict avoidance)

__global__ __launch_bounds__(256)
void ssm_rk4_wmma_kernel(const float* __restrict__ x,
                         const float* __restrict__ state,
                         const float* __restrict__ A,
                         const float* __restrict__ B,
                         const float* __restrict__ C,
                         const float* __restrict__ D,
                         const float* __restrict__ dtp,
                         float* __restrict__ y_out,
                         float* __restrict__ state_out)
{
    __shared__ float sA[SSM_DTILE * SSM_LSTRIDE];
    __shared__ float sB[SSM_DTILE * SSM_LSTRIDE];
    __shared__ float sC[SSM_DTILE * SSM_LSTRIDE];
    __shared__ float sdt[SSM_DTILE];
    __shared__ float ybounce[SSM_BWAVES][SSM_DTILE];

    const int tid = threadIdx.x;
    const int d0  = blockIdx.x * SSM_DTILE;
    const int b0  = blockIdx.y * SSM_BWAVES;

    // ---- Stage A/B/C tile (16 d x 16 n each) into LDS, coalesced ----
    {
        const int dd = tid >> 4;          // 0..15
        const int nn = tid & 15;          // 0..15
        const int g  = (d0 + dd) * SSM_DSTATE + nn;
        const int l  = dd * SSM_LSTRIDE + nn;
        sA[l] = A[g];
        sB[l] = B[g];
        sC[l] = C[g];
    }
    if (tid < SSM_DTILE) {
        // dt = softplus(dt_proj); dt_proj in [1e-3, 1e-1] so no overflow care
        const float v = dtp[d0 + tid];
        sdt[tid] = log1pf(expf(v));
    }
    __syncthreads();

    // ---- Lane mapping: wave = batch, lane = (d_local, n-half) ----
    const int wave = tid >> 5;            // 0..7 -> batch within block
    const int lane = tid & 31;
    const int dl   = lane & 15;           // d_local == WMMA A-matrix row M
    const int half = lane >> 4;           // n in [half*8, half*8+8)
    const int b    = b0 + wave;
    const int d    = d0 + dl;

    const float xv = x[(size_t)b * SSM_DINNER + d];
    const float dt = sdt[dl];
    const float Dx = D[d] * xv;

    // ---- Stream state half-row (8 floats, 2x b128) + fused passthrough copy ----
    const size_t rowoff = ((size_t)b * SSM_DINNER + d) * SSM_DSTATE + half * 8;
    const v4f s0 = *(const v4f*)(state + rowoff);
    const v4f s1 = *(const v4f*)(state + rowoff + 4);
    *(v4f*)(state_out + rowoff)     = s0;
    *(v4f*)(state_out + rowoff + 4) = s1;

    // ---- RK4 (classic stages, fp32 FMA) and fold C: t = new_state * C ----
    const int   lbase = dl * SSM_LSTRIDE + half * 8;
    const float h  = dt;
    const float hh = 0.5f * dt;
    const float h6 = dt * (1.0f / 6.0f);

    float t[8];
#pragma unroll
    for (int i = 0; i < 8; ++i) {
        const float s  = (i < 4) ? s0[i] : s1[i - 4];
        const float a  = sA[lbase + i];
        const float bx = sB[lbase + i] * xv;
        const float k1 = fmaf(a, s, bx);
        const float k2 = fmaf(a, fmaf(hh, k1, s), bx);
        const float k3 = fmaf(a, fmaf(hh, k2, s), bx);
        const float k4 = fmaf(a, fmaf(h,  k3, s), bx);
        const float ks = k1 + 2.0f * (k2 + k3) + k4;
        const float ns = fmaf(h6, ks, s);
        t[i] = ns * sC[lbase + i];
    }

    // ---- Row-sum over n via V_WMMA_F32_16X16X4_F32 against all-ones B ----
    // A layout (f32 16x4): lanes 0-15 = rows M, k0/k1; lanes 16-31 = rows, k2/k3.
    // Our lane holds n = half*8 + i, so wmma #w consumes (t[2w], t[2w+1]):
    //   low lanes  -> k = {2w, 2w+1}   == n {2w, 2w+1}
    //   high lanes -> k = {2w+2?No: k2,k3} == n {8+2w, 8+2w+1}
    // Over w=0..3 every n in 0..15 is summed exactly once per row.
    v8f acc;
#pragma unroll
    for (int i = 0; i < 8; ++i) acc[i] = 0.0f;
    v2f ones; ones[0] = 1.0f; ones[1] = 1.0f;

#pragma unroll
    for (int w = 0; w < 4; ++w) {
        v2f a2; a2[0] = t[2 * w]; a2[1] = t[2 * w + 1];
        acc = __builtin_amdgcn_wmma_f32_16x16x4_f32(
            false, a2, false, ones, (short)0, acc, false, false);
    }

    // All columns of acc are identical (B = ones). Lane 0 holds y-rows 0..7 in
    // acc[0..7], lane 16 holds rows 8..15. Bounce through LDS (intra-wave; the
    // compiler orders ds store->load with s_wait_dscnt), then coalesced store.
    if (dl == 0) {
        const int base = half * 8;
#pragma unroll
        for (int i = 0; i < 8; ++i) ybounce[wave][base + i] = acc[i];
    }
    if (lane < 16) {
        const float yv = ybounce[wave][lane] + Dx;
        y_out[(size_t)b * SSM_DINNER + d] = yv;
    }
}

extern "C" void kernel_launch(void* const* d_in, const int* in_sizes, int n_in,
                              void* d_out, int out_size, void* d_ws, size_t ws_size,
                              hipStream_t stream) {
    (void)in_sizes; (void)n_in; (void)out_size; (void)d_ws; (void)ws_size;
    const float* x     = (const float*)d_in[0];
    const float* state = (const float*)d_in[1];
    const float* A     = (const float*)d_in[2];
    const float* B     = (const float*)d_in[3];
    const float* C     = (const float*)d_in[4];
    const float* D     = (const float*)d_in[5];
    const float* dtp   = (const float*)d_in[6];

    float* y_out     = (float*)d_out;
    float* state_out = y_out + (size_t)SSM_BATCH * SSM_DINNER;

    dim3 grid(SSM_DINNER / SSM_DTILE, SSM_BATCH / SSM_BWAVES);  // (256, 128)
    dim3 block(256);
    ssm_rk4_wmma_kernel<<<grid, block, 0, stream>>>(
        x, state, A, B, C, D, dtp, y_out, state_out);
}